// ESN_51101520888403
// MI455X (gfx1250) — compile-verified
//
#include <hip/hip_runtime.h>
#include <math.h>

// ESN forward for MI455X (gfx1250, wave32).
// B=128, T=1024, D=8, R=512.
//
// u = x @ W_in^T computed in-LDS with v_wmma_f32_16x16x4_f32 (K=8 as 2x K=4).
// Scan: each wave redundantly evolves a 64-wide reservoir window (32 owned
// columns + 16-halo each side, 2 values/lane) for 16 steps using only two
// intra-wave shuffles per step -> no barriers / no LDS in the sequential
// chain. Halo garbage travels 1 slot/step and never reaches the owned region.
// State handoff between chunks via ping-ponged LDS (2 barriers per 16 steps).
// tanh uses gfx1250 hardware v_tanh_f32 when available.
// Output (268 MB > 192 MB L2) uses non-temporal streaming stores.

#define B_ 128
#define T_ 1024
#define D_ 8
#define R_ 512
#define CH 16               // timesteps per chunk (= halo width)
#define NCHUNK (T_ / CH)

typedef __attribute__((ext_vector_type(2))) float v2f;
typedef __attribute__((ext_vector_type(8))) float v8f;

__device__ __forceinline__ float tanh_fast(float v) {
#if __has_builtin(__builtin_amdgcn_tanhf)
    return __builtin_amdgcn_tanhf(v);          // v_tanh_f32
#elif __has_builtin(__builtin_amdgcn_tanh_f32)
    return __builtin_amdgcn_tanh_f32(v);
#else
    return tanhf(v);
#endif
}

__global__ __launch_bounds__(R_) void ESN_51101520888403_kernel(
    const float* __restrict__ x,      // [B, T, D]
    const float* __restrict__ W_in,   // [R, D]
    float* __restrict__ out)          // [B, T, R]
{
    __shared__ float u_buf[CH * R_];   // 32 KB: u for 16 timesteps
    __shared__ float stbuf[2 * R_];    // 4 KB: ping-pong chunk-boundary state

    const int tid  = threadIdx.x;      // 0..511
    const int lane = tid & 31;
    const int wave = tid >> 5;         // 0..15
    const int half = lane >> 4;        // 0/1
    const int mrow = lane & 15;        // M (or N) index within WMMA tile
    const int b    = blockIdx.x;

    const float* xb   = x + (size_t)b * T_ * D_;
    float*       outb = out + (size_t)b * T_ * R_;

    // ---- Preload B-matrix fragments (W_in^T), invariant across chunks ----
    // Wave w owns N-tiles {2w, 2w+1}. B 4x16 f32: lane l -> N=l%16, K=v+2*(l/16).
    v2f bf[2][2]; // [ntile][kslice]
#pragma unroll
    for (int nt = 0; nt < 2; ++nt) {
        const int rcol = (wave * 2 + nt) * 16 + mrow;
#pragma unroll
        for (int ks = 0; ks < 2; ++ks) {
            const int k = ks * 4 + 2 * half;
            bf[nt][ks].x = W_in[rcol * D_ + k];
            bf[nt][ks].y = W_in[rcol * D_ + k + 1];
        }
    }

    // ---- Scan window geometry ----
    // Wave w owns reservoir columns [32w, 32w+32). Window = [32w-16, 32w+48).
    // Lane l holds window positions 2l, 2l+1 -> global g0 = (32w-16+2l) mod 512.
    const int  g0    = (wave * 32 - 16 + 2 * lane) & (R_ - 1);   // even
    const bool owned = (lane >= 8) && (lane < 24);               // local [16,48)

    // ---- Init chunk-boundary state to zero ----
    stbuf[tid]      = 0.0f;
    stbuf[R_ + tid] = 0.0f;
    int p = 0;

    for (int c = 0; c < NCHUNK; ++c) {
        const int t0 = c * CH;

        // ---- Phase 1: u[t0..t0+15][:] into LDS via WMMA f32 16x16x4 ----
        // A 16x4 f32: lane l -> M=l%16, K=v+2*(l/16).
        v2f af[2];
#pragma unroll
        for (int ks = 0; ks < 2; ++ks) {
            const int k = ks * 4 + 2 * half;
            const float* xr = xb + (size_t)(t0 + mrow) * D_ + k;
            af[ks].x = xr[0];
            af[ks].y = xr[1];
        }
#pragma unroll
        for (int nt = 0; nt < 2; ++nt) {
            v8f acc = {};
            acc = __builtin_amdgcn_wmma_f32_16x16x4_f32(
                false, af[0], false, bf[nt][0], (short)0, acc, false, false);
            acc = __builtin_amdgcn_wmma_f32_16x16x4_f32(
                false, af[1], false, bf[nt][1], (short)0, acc, false, false);
            // C 16x16 f32: lane l -> M = v + 8*(l/16), N = l%16
            const int ncol = (wave * 2 + nt) * 16 + mrow;
#pragma unroll
            for (int v = 0; v < 8; ++v) {
                const int m = v + 8 * half;
                u_buf[m * R_ + ncol] = acc[v];
            }
        }
        __syncthreads();   // u_buf ready; prev chunk's state stores visible

        // ---- Load state window + preload all 16 u pairs into registers ----
        v2f s = *(const v2f*)&stbuf[p * R_ + g0];
        v2f uv[CH];
#pragma unroll
        for (int m = 0; m < CH; ++m)
            uv[m] = *(const v2f*)&u_buf[m * R_ + g0];

        // ---- Phase 2: 16 shuffle-only scan steps ----
        // new[j] = tanh(0.15*old[j+1] + 0.85*old[j-1] + u[j])
#pragma unroll
        for (int m = 0; m < CH; ++m) {
            if (c == 0 && m == 0) continue;   // reference skips u[:,0,:]
            const float s1u = __shfl(s.y, lane - 1);  // old[2l-1]
            const float s0d = __shfl(s.x, lane + 1);  // old[2l+2]
            const float n0 = tanh_fast(0.15f * s.y + 0.85f * s1u + uv[m].x);
            const float n1 = tanh_fast(0.15f * s0d + 0.85f * s.x + uv[m].y);
            s.x = n0;
            s.y = n1;
            if (owned) {
                v2f o; o.x = n0; o.y = n1;
                __builtin_nontemporal_store(
                    o, (v2f*)&outb[(size_t)(t0 + m - 1) * R_ + g0]);
            }
        }

        // ---- Hand final owned state to the other ping-pong buffer ----
        if (owned)
            *(v2f*)&stbuf[(p ^ 1) * R_ + g0] = s;
        __syncthreads();   // state visible; u_buf reads done before next WMMA
        p ^= 1;
    }

    // ---- Zero tail: out[b, T-1, :] = 0 ----
    {
        float z = 0.0f;
        __builtin_nontemporal_store(z, &outb[(size_t)(T_ - 1) * R_ + tid]);
    }
}

extern "C" void kernel_launch(void* const* d_in, const int* in_sizes, int n_in,
                              void* d_out, int out_size, void* d_ws, size_t ws_size,
                              hipStream_t stream)
{
    (void)in_sizes; (void)n_in; (void)out_size; (void)d_ws; (void)ws_size;
    const float* x    = (const float*)d_in[0];
    const float* W_in = (const float*)d_in[1];
    float*       out  = (float*)d_out;

    ESN_51101520888403_kernel<<<B_, R_, 0, stream>>>(x, W_in, out);
}